// RefLocal_58334245814786
// MI455X (gfx1250) — compile-verified
//
#include <hip/hip_runtime.h>
#include <hip/hip_bf16.h>

// Windowed local 7x7 attention, H=W=256, C=32, fp32.
// Per-wave 16-pixel row tile; both contractions on V_WMMA_F32_16X16X4_F32:
//   S = Main(16x32) * RefCols(32x32)  per dy  (scores = diagonal band)
//   O += Wband(16x32) * ValCols(32x32) per dy (softmax-weighted sum)
// OOB handling is branchless: clamp address, always load, cndmask to zero.

typedef float v2f __attribute__((ext_vector_type(2)));
typedef float v8f __attribute__((ext_vector_type(8)));

#define Hh 256
#define Ww 256
#define Cc 32
#define RAD 3
#define KW 7
#define WAVES_PER_BLOCK 4
#define SC_FLOATS (KW * 16 * 32)
#define W_FLOATS  (16 * 32)
#define WAVE_LDS  (SC_FLOATS + W_FLOATS)

__global__ __launch_bounds__(WAVES_PER_BLOCK * 32)
void local_attn_wmma(const float* __restrict__ mainp,
                     const float* __restrict__ refp,
                     const float* __restrict__ valp,
                     float* __restrict__ outp) {
    __shared__ float lds[WAVES_PER_BLOCK * WAVE_LDS];

    const int lane = threadIdx.x & 31;
    const int wv   = threadIdx.x >> 5;
    float* sc   = lds + wv * WAVE_LDS;      // [7][16][32] scores
    float* wmat = sc + SC_FLOATS;           // [16][32] banded softmax weights

    const int tile = blockIdx.x * WAVES_PER_BLOCK + wv;  // 4096 tiles total
    const int h  = tile >> 4;            // row
    const int w0 = (tile & 15) << 4;     // first pixel column of tile
    const int j0 = w0 - 8;               // left edge of 32-wide column window

    const int  ml = lane & 15;           // M index (A frags), N index (B/C/D frags)
    const bool hi = lane >= 16;          // upper half: K offset +2 (A/B), M offset +8 (C/D)
    const int  kofs = hi ? 2 : 0;

    // ---------------- A fragments: main[h, w0+ml, k], k = 4*ch + kofs + {0,1}
    v2f a[8];
    {
        const float* mrow = mainp + ((size_t)h * Ww + (w0 + ml)) * Cc + kofs;
        #pragma unroll
        for (int ch = 0; ch < 8; ++ch)
            a[ch] = *(const v2f*)(mrow + 4 * ch);
    }

    // ---------------- Stage 1: scores via WMMA, store to LDS
    for (int dy = 0; dy < KW; ++dy) {
        const int  r   = h + dy - RAD;
        const bool rok = (r >= 0) & (r < Hh);
        const int  rc  = rok ? r : 0;                    // clamped (always valid)
        #pragma unroll
        for (int t = 0; t < 2; ++t) {
            const int  n   = j0 + 16 * t + ml;           // global ref column
            const bool nok = rok & (n >= 0) & (n < Ww);
            const int  nc  = (n < 0) ? 0 : ((n >= Ww) ? (Ww - 1) : n);
            const float* bcol = refp + ((size_t)rc * Ww + nc) * Cc + kofs;

            // Batch all 8 fragment loads (branchless; zero via select after load)
            v2f b[8];
            #pragma unroll
            for (int ch = 0; ch < 8; ++ch)
                b[ch] = *(const v2f*)(bcol + 4 * ch);
            const v2f z2 = {0.f, 0.f};
            #pragma unroll
            for (int ch = 0; ch < 8; ++ch)
                b[ch] = nok ? b[ch] : z2;

            v8f acc = {0.f, 0.f, 0.f, 0.f, 0.f, 0.f, 0.f, 0.f};
            #pragma unroll
            for (int ch = 0; ch < 8; ++ch)
                acc = __builtin_amdgcn_wmma_f32_16x16x4_f32(
                        false, a[ch], false, b[ch], (short)0, acc, false, false);

            // C/D layout: VGPR v -> M = v + (hi?8:0), N = ml (+16t)
            #pragma unroll
            for (int v = 0; v < 8; ++v) {
                const int m = v + (hi ? 8 : 0);
                sc[(dy * 16 + m) * 32 + (ml + 16 * t)] = acc[v];
            }
        }
    }
    __syncthreads();

    // ---------------- Softmax stats per pixel (band: jloc = m + dxi + 5)
    float mx = -3.0e38f;
    #pragma unroll
    for (int dy = 0; dy < KW; ++dy)
        #pragma unroll
        for (int dxi = 0; dxi < KW; ++dxi)
            mx = fmaxf(mx, sc[(dy * 16 + ml) * 32 + (ml + dxi + 5)]);
    float sum = 0.f;
    #pragma unroll
    for (int dy = 0; dy < KW; ++dy)
        #pragma unroll
        for (int dxi = 0; dxi < KW; ++dxi)
            sum += __expf(sc[(dy * 16 + ml) * 32 + (ml + dxi + 5)] - mx);
    const float rsum = 1.0f / sum;

    // ---------------- Stage 2: out += Wband * Val, per dy
    v8f o0 = {0.f, 0.f, 0.f, 0.f, 0.f, 0.f, 0.f, 0.f};
    v8f o1 = {0.f, 0.f, 0.f, 0.f, 0.f, 0.f, 0.f, 0.f};
    for (int dy = 0; dy < KW; ++dy) {
        const int  r   = h + dy - RAD;
        const bool rok = (r >= 0) & (r < Hh);
        const int  rc  = rok ? r : 0;
        const float* vrow = valp + (size_t)rc * Ww * Cc;

        __syncthreads();                       // protect previous iter's wmat reads
        for (int ii = lane; ii < W_FLOATS; ii += 32) wmat[ii] = 0.f;
        if (lane < 16) {
            #pragma unroll
            for (int dxi = 0; dxi < KW; ++dxi) {
                const float s = sc[(dy * 16 + ml) * 32 + (ml + dxi + 5)];
                wmat[ml * 32 + (ml + dxi + 5)] = __expf(s - mx) * rsum;
            }
        }
        __syncthreads();

        // A2 fragments from wmat: M = ml, K = 4*ch + kofs + {0,1}
        v2f a2[8];
        #pragma unroll
        for (int ch = 0; ch < 8; ++ch)
            a2[ch] = *(const v2f*)(wmat + ml * 32 + 4 * ch + kofs);

        // Batch ALL val fragment loads for both channel tiles (branchless)
        v2f b2[2][8];
        #pragma unroll
        for (int t = 0; t < 2; ++t) {
            const int c = ml + 16 * t;         // output channel (N index)
            #pragma unroll
            for (int ch = 0; ch < 8; ++ch) {
                const int jA  = j0 + 4 * ch + kofs;   // K = window column index
                const int jB  = jA + 1;
                const int jAc = (jA < 0) ? 0 : ((jA >= Ww) ? (Ww - 1) : jA);
                const int jBc = (jB < 0) ? 0 : ((jB >= Ww) ? (Ww - 1) : jB);
                v2f b;
                b.x = vrow[(size_t)jAc * Cc + c];
                b.y = vrow[(size_t)jBc * Cc + c];
                b.x = (rok & (jA >= 0) & (jA < Ww)) ? b.x : 0.f;
                b.y = (rok & (jB >= 0) & (jB < Ww)) ? b.y : 0.f;
                b2[t][ch] = b;
            }
        }

        #pragma unroll
        for (int ch = 0; ch < 8; ++ch)
            o0 = __builtin_amdgcn_wmma_f32_16x16x4_f32(
                    false, a2[ch], false, b2[0][ch], (short)0, o0, false, false);
        #pragma unroll
        for (int ch = 0; ch < 8; ++ch)
            o1 = __builtin_amdgcn_wmma_f32_16x16x4_f32(
                    false, a2[ch], false, b2[1][ch], (short)0, o1, false, false);
    }

    // ---------------- Store: D layout VGPR v -> M = v + (hi?8:0), N = channel
    #pragma unroll
    for (int v = 0; v < 8; ++v) {
        const int m = v + (hi ? 8 : 0);
        float* orow = outp + ((size_t)h * Ww + (w0 + m)) * Cc;
        orow[ml]      = o0[v];
        orow[ml + 16] = o1[v];
    }
}

extern "C" void kernel_launch(void* const* d_in, const int* in_sizes, int n_in,
                              void* d_out, int out_size, void* d_ws, size_t ws_size,
                              hipStream_t stream) {
    (void)in_sizes; (void)n_in; (void)d_ws; (void)ws_size; (void)out_size;
    const float* mainp = (const float*)d_in[0];
    const float* refp  = (const float*)d_in[1];
    const float* valp  = (const float*)d_in[2];
    // d_in[3] = k_size (==7), baked in as a compile-time constant.
    float* outp = (float*)d_out;

    const int tiles = Hh * (Ww / 16);                    // 4096 16-pixel tiles
    dim3 grid(tiles / WAVES_PER_BLOCK), block(WAVES_PER_BLOCK * 32);
    local_attn_wmma<<<grid, block, 0, stream>>>(mainp, refp, valp, outp);
}